// GraphConvLayer_3607772529056
// MI455X (gfx1250) — compile-verified
//
#include <hip/hip_runtime.h>
#include <stdint.h>

#define IN_CH 128
#define OUT_CH 64

typedef __attribute__((ext_vector_type(2))) float v2f;
typedef __attribute__((ext_vector_type(4))) float v4f;
typedef __attribute__((ext_vector_type(8))) float v8f;

// ---------------------------------------------------------------------------
// Kernel 1: xw = x @ W using V_WMMA_F32_16X16X4_F32.
// One wave computes a 16x64 tile (4 accumulators of 16x16), K=128 in 32 steps.
// A layout (fp32 16x4): lane = 16*khalf + m ; VGPR0/1 = K = 2*khalf + {0,1}
// B layout (fp32 4x16): lane = 16*khalf + n ; VGPR0/1 = K = 2*khalf + {0,1}
// C/D layout: VGPR r -> M = r + 8*(lane>>4), N = lane&15
// ---------------------------------------------------------------------------
__global__ __launch_bounds__(256) void gcn_gemm_wmma(
    const float* __restrict__ x, const float* __restrict__ W,
    float* __restrict__ xw, int n_nodes) {
  const int lane = threadIdx.x & 31;
  const int wave = threadIdx.x >> 5;
  const int tile = blockIdx.x * 8 + wave;          // 16-row tile index
  if (tile * 16 >= n_nodes) return;                // wave-uniform exit: EXEC all-1s in WMMA

  const int m  = lane & 15;
  const int kb = (lane >> 4) * 2;                  // 0 or 2 (K half select)
  const float* __restrict__ xrow = x + (size_t)(tile * 16 + m) * IN_CH;

  v8f acc0 = {}, acc1 = {}, acc2 = {}, acc3 = {};

  for (int k = 0; k < IN_CH; k += 4) {
    // A: 16x4 fp32 tile, 2 VGPRs per lane (contiguous in K -> one b64 load)
    v2f a = *(const v2f*)(xrow + k + kb);

    // B: 4x16 fp32 tiles for the 4 column groups (n = lane&15)
    const float* __restrict__ w0 = W + (size_t)(k + kb) * OUT_CH;       // K row kb
    const float* __restrict__ w1 = W + (size_t)(k + kb + 1) * OUT_CH;   // K row kb+1
    v2f b0, b1, b2, b3;
    b0.x = w0[m];      b0.y = w1[m];
    b1.x = w0[16 + m]; b1.y = w1[16 + m];
    b2.x = w0[32 + m]; b2.y = w1[32 + m];
    b3.x = w0[48 + m]; b3.y = w1[48 + m];

    acc0 = __builtin_amdgcn_wmma_f32_16x16x4_f32(false, a, false, b0, (short)0, acc0, false, false);
    acc1 = __builtin_amdgcn_wmma_f32_16x16x4_f32(false, a, false, b1, (short)0, acc1, false, false);
    acc2 = __builtin_amdgcn_wmma_f32_16x16x4_f32(false, a, false, b2, (short)0, acc2, false, false);
    acc3 = __builtin_amdgcn_wmma_f32_16x16x4_f32(false, a, false, b3, (short)0, acc3, false, false);
  }

  // Store D: VGPR r holds row M = r + 8*(lane>>4), column N = lane&15
  const int mBase = tile * 16 + (lane >> 4) * 8;
  const int n = lane & 15;
#pragma unroll
  for (int r = 0; r < 8; ++r) {
    float* __restrict__ orow = xw + (size_t)(mBase + r) * OUT_CH;
    orow[n]      = acc0[r];
    orow[16 + n] = acc1[r];
    orow[32 + n] = acc2[r];
    orow[48 + n] = acc3[r];
  }
}

// ---------------------------------------------------------------------------
// Kernel 2: deg[i] = 1 (self loop)
// ---------------------------------------------------------------------------
__global__ void gcn_deg_init(uint32_t* __restrict__ deg, int n) {
  int i = blockIdx.x * blockDim.x + threadIdx.x;
  if (i < n) deg[i] = 1u;
}

// ---------------------------------------------------------------------------
// Kernel 3: integer degree histogram over target nodes (exact, deterministic)
// ---------------------------------------------------------------------------
__global__ void gcn_deg_accum(const long long* __restrict__ ei,
                              uint32_t* __restrict__ deg, int nE) {
  int e = blockIdx.x * blockDim.x + threadIdx.x;
  if (e < nE) {
    int col = (int)ei[(size_t)nE + e];
    atomicAdd(&deg[col], 1u);
  }
}

// ---------------------------------------------------------------------------
// Kernel 4: dinv[i] = rsqrt(deg[i])  (deg >= 1 always due to self loop)
// ---------------------------------------------------------------------------
__global__ void gcn_dinv(const uint32_t* __restrict__ deg,
                         float* __restrict__ dinv, int n) {
  int i = blockIdx.x * blockDim.x + threadIdx.x;
  if (i < n) dinv[i] = rsqrtf((float)deg[i]);
}

// ---------------------------------------------------------------------------
// Kernel 5: out[i][:] = b + xw[i][:] * dinv[i]^2   (self-loop msg + bias)
// 16 threads per node, float4 per thread. Non-atomic init before scatter.
// ---------------------------------------------------------------------------
__global__ void gcn_out_init(const float* __restrict__ xw,
                             const float* __restrict__ dinv,
                             const float* __restrict__ b,
                             float* __restrict__ out, int n) {
  int t = blockIdx.x * blockDim.x + threadIdx.x;
  int i = t >> 4;
  int c = (t & 15) << 2;
  if (i >= n) return;
  float di = dinv[i];
  float w = di * di;
  v4f mv = *(const v4f*)(xw + (size_t)i * OUT_CH + c);
  v4f bv = *(const v4f*)(b + c);
  v4f r;
  r.x = bv.x + mv.x * w;
  r.y = bv.y + mv.y * w;
  r.z = bv.z + mv.z * w;
  r.w = bv.w + mv.w * w;
  *(v4f*)(out + (size_t)i * OUT_CH + c) = r;
}

// ---------------------------------------------------------------------------
// Kernel 6: edge scatter. 16 lanes per edge; float4 gather from xw[row],
// scale by dinv[row]*dinv[col], f32 atomic adds into out[col] (L2-resident).
// ---------------------------------------------------------------------------
__global__ __launch_bounds__(256) void gcn_edge_scatter(
    const long long* __restrict__ ei, const float* __restrict__ xw,
    const float* __restrict__ dinv, float* __restrict__ out, int nE) {
  int t = blockIdx.x * blockDim.x + threadIdx.x;
  int e = t >> 4;
  int c = (t & 15) << 2;
  if (e >= nE) return;
  int row = (int)ei[e];
  int col = (int)ei[(size_t)nE + e];
  float w = dinv[row] * dinv[col];
  v4f mv = *(const v4f*)(xw + (size_t)row * OUT_CH + c);
  float* __restrict__ o = out + (size_t)col * OUT_CH + c;
  atomicAdd(o + 0, mv.x * w);
  atomicAdd(o + 1, mv.y * w);
  atomicAdd(o + 2, mv.z * w);
  atomicAdd(o + 3, mv.w * w);
}

extern "C" void kernel_launch(void* const* d_in, const int* in_sizes, int n_in,
                              void* d_out, int out_size, void* d_ws, size_t ws_size,
                              hipStream_t stream) {
  const float*     x  = (const float*)d_in[0];
  const long long* ei = (const long long*)d_in[1];   // int64 edge_index [2, E]
  const float*     W  = (const float*)d_in[2];
  const float*     b  = (const float*)d_in[3];
  float* out = (float*)d_out;

  const int N = in_sizes[0] / IN_CH;   // 100000
  const int E = in_sizes[1] / 2;       // 1600000

  // Workspace layout: xw [N*64 f32] | deg [N u32] | dinv [N f32]
  char* ws = (char*)d_ws;
  float* xw = (float*)ws;
  size_t xwB = ((size_t)N * OUT_CH * sizeof(float) + 255) & ~(size_t)255;
  uint32_t* deg = (uint32_t*)(ws + xwB);
  size_t degB = ((size_t)N * sizeof(uint32_t) + 255) & ~(size_t)255;
  float* dinv = (float*)(ws + xwB + degB);

  const int tiles = (N + 15) / 16;  // 6250, exact
  gcn_gemm_wmma<<<(tiles + 7) / 8, 256, 0, stream>>>(x, W, xw, N);
  gcn_deg_init<<<(N + 255) / 256, 256, 0, stream>>>(deg, N);
  gcn_deg_accum<<<(E + 255) / 256, 256, 0, stream>>>(ei, deg, E);
  gcn_dinv<<<(N + 255) / 256, 256, 0, stream>>>(deg, dinv, N);
  gcn_out_init<<<(N * 16 + 255) / 256, 256, 0, stream>>>(xw, dinv, b, out, N);
  gcn_edge_scatter<<<((size_t)E * 16 + 255) / 256, 256, 0, stream>>>(ei, xw, dinv, out, E);
}